// SAGE_7404523618676
// MI455X (gfx1250) — compile-verified
//
#include <hip/hip_runtime.h>

#define N_NODES 100000
#define N_EDGES 800000
#define IN_DIM  128
#define HID_DIM 256
#define OUT_DIM 128

typedef float v2f __attribute__((ext_vector_type(2)));
typedef float v8f __attribute__((ext_vector_type(8)));

// D = A(16x4 f32) * B(4x16 f32) + C(16x16 f32), wave32, one tile per wave.
__device__ __forceinline__ v8f wmma_f32_16x16x4(v2f a, v2f b, v8f c) {
    return __builtin_amdgcn_wmma_f32_16x16x4_f32(
        /*neg_a=*/false, a, /*neg_b=*/false, b,
        /*c_mod=*/(short)0, c, /*reuse_a=*/false, /*reuse_b=*/false);
}

// ---------------------------------------------------------------------------
// Edge scatter-add: one lane per (edge, 4-float chunk). Consecutive lanes
// cover consecutive chunks of the same edge -> coalesced 16B gathers from the
// (L2-resident) feature matrix, 4x global_atomic_add_f32 scatter.
// ---------------------------------------------------------------------------
__global__ void sage_scatter_kernel(const float* __restrict__ feat,
                                    const int*   __restrict__ src,
                                    const int*   __restrict__ dst,
                                    float* __restrict__ agg,
                                    float* __restrict__ cnt,   // null for layer 2
                                    int n_edges, int dim) {
    const int chunks = dim >> 2;
    const long long tid   = (long long)blockIdx.x * blockDim.x + threadIdx.x;
    const long long total = (long long)n_edges * chunks;
    if (tid >= total) return;
    const int e = (int)(tid / chunks);
    const int c = (int)(tid % chunks);
    const int s = src[e];
    const int d = dst[e];
    const float4 v = *(const float4*)(feat + (long long)s * dim + c * 4);
    float* o = agg + (long long)d * dim + c * 4;
    atomicAdd(o + 0, v.x);
    atomicAdd(o + 1, v.y);
    atomicAdd(o + 2, v.z);
    atomicAdd(o + 3, v.w);
    if (cnt && c == 0) atomicAdd(cnt + d, 1.0f);
}

__global__ void sage_invcnt_kernel(const float* __restrict__ cnt,
                                   float* __restrict__ inv, int n) {
    const int i = blockIdx.x * blockDim.x + threadIdx.x;
    if (i < n) inv[i] = 1.0f / fmaxf(cnt[i], 1.0f);
}

// ---------------------------------------------------------------------------
// Fused SAGE GEMM: out[16x(16*NT)] tile per wave.
//   out = act( (agg .* inv_cnt) @ Wl^T + bias + xself @ Wr^T )
// A fragments loaded per K-step (mean-scaling fused into the A load), two
// WMMAs per N-tile per K-step accumulate into the same v8f accumulator.
// A frag layout : lanes 0-15 hold row M=lane, K={k,k+1}; lanes 16-31 K={k+2,k+3}
// B frag layout : lanes 0-15 hold col N=lane, K={k,k+1}; lanes 16-31 K={k+2,k+3}
// C/D layout    : VGPR j -> M = j + 8*(lane>>4), N = lane&15
// ---------------------------------------------------------------------------
template <int K_DIM, int N_OUT, int NT, bool RELU>
__global__ void sage_gemm_kernel(const float* __restrict__ agg,
                                 const float* __restrict__ inv_cnt,
                                 const float* __restrict__ xself,
                                 const float* __restrict__ Wl,   // [N_OUT][K_DIM]
                                 const float* __restrict__ Wr,   // [N_OUT][K_DIM]
                                 const float* __restrict__ bias, // [N_OUT]
                                 float* __restrict__ out,        // [n][N_OUT]
                                 int n_nodes) {
    constexpr int N_GROUPS = N_OUT / (16 * NT);
    const int lane  = threadIdx.x & 31;
    const int wave  = (blockIdx.x * blockDim.x + threadIdx.x) >> 5;
    const int mtile = wave / N_GROUPS;
    const int ngrp  = wave % N_GROUPS;
    if (mtile * 16 >= n_nodes) return;           // wave-uniform guard (EXEC all-1s)

    const int row  = mtile * 16 + (lane & 15);
    const int koff = (lane >> 4) * 2;            // upper half-wave holds K+2,K+3
    const float inv = inv_cnt[row];

    const float* arow = agg   + (long long)row * K_DIM + koff;
    const float* xrow = xself + (long long)row * K_DIM + koff;

    v8f acc[NT];
#pragma unroll
    for (int t = 0; t < NT; ++t)
#pragma unroll
        for (int j = 0; j < 8; ++j) acc[t][j] = 0.0f;

    int ncol[NT];
#pragma unroll
    for (int t = 0; t < NT; ++t) ncol[t] = (ngrp * NT + t) * 16 + (lane & 15);

#pragma unroll 4
    for (int k = 0; k < K_DIM; k += 4) {
        const float2 am = *(const float2*)(arow + k);
        const float2 as = *(const float2*)(xrow + k);
        v2f a_mean; a_mean.x = am.x * inv; a_mean.y = am.y * inv;
        v2f a_self; a_self.x = as.x;       a_self.y = as.y;
#pragma unroll
        for (int t = 0; t < NT; ++t) {
            const float2 wl = *(const float2*)(Wl + (long long)ncol[t] * K_DIM + k + koff);
            const float2 wr = *(const float2*)(Wr + (long long)ncol[t] * K_DIM + k + koff);
            v2f bl; bl.x = wl.x; bl.y = wl.y;
            v2f br; br.x = wr.x; br.y = wr.y;
            acc[t] = wmma_f32_16x16x4(a_mean, bl, acc[t]);
            acc[t] = wmma_f32_16x16x4(a_self, br, acc[t]);
        }
    }

#pragma unroll
    for (int t = 0; t < NT; ++t) {
        const float bv = bias[ncol[t]];
#pragma unroll
        for (int j = 0; j < 8; ++j) {
            const int r = mtile * 16 + (lane >> 4) * 8 + j;
            float v = acc[t][j] + bv;
            if (RELU) v = fmaxf(v, 0.0f);
            out[(long long)r * N_OUT + ncol[t]] = v;
        }
    }
}

// ---------------------------------------------------------------------------
extern "C" void kernel_launch(void* const* d_in, const int* in_sizes, int n_in,
                              void* d_out, int out_size, void* d_ws, size_t ws_size,
                              hipStream_t stream) {
    const float* x    = (const float*)d_in[0];
    const int*   edge = (const int*)  d_in[1];   // [2, E]
    const float* W1l  = (const float*)d_in[2];
    const float* b1   = (const float*)d_in[3];
    const float* W1r  = (const float*)d_in[4];
    const float* W2l  = (const float*)d_in[5];
    const float* b2   = (const float*)d_in[6];
    const float* W2r  = (const float*)d_in[7];
    const int* src = edge;
    const int* dst = edge + N_EDGES;
    float* out = (float*)d_out;

    // Workspace carve-out (~257 MB): agg1, h, agg2, cnt, inv
    char* ws = (char*)d_ws;
    size_t off = 0;
    auto carve = [&](size_t bytes) -> void* {
        void* p = ws + off;
        off += (bytes + 255) & ~(size_t)255;
        return p;
    };
    float* agg1 = (float*)carve(sizeof(float) * (size_t)N_NODES * IN_DIM);
    float* h    = (float*)carve(sizeof(float) * (size_t)N_NODES * HID_DIM);
    float* agg2 = (float*)carve(sizeof(float) * (size_t)N_NODES * HID_DIM);
    float* cnt  = (float*)carve(sizeof(float) * (size_t)N_NODES);
    float* inv  = (float*)carve(sizeof(float) * (size_t)N_NODES);

    hipMemsetAsync(agg1, 0, sizeof(float) * (size_t)N_NODES * IN_DIM,  stream);
    hipMemsetAsync(agg2, 0, sizeof(float) * (size_t)N_NODES * HID_DIM, stream);
    hipMemsetAsync(cnt,  0, sizeof(float) * (size_t)N_NODES,           stream);

    // Layer 1: scatter-mean of x, then h = relu(mean@W1l^T + b1 + x@W1r^T)
    {
        const long long total = (long long)N_EDGES * (IN_DIM / 4);
        const int blocks = (int)((total + 255) / 256);
        sage_scatter_kernel<<<blocks, 256, 0, stream>>>(x, src, dst, agg1, cnt,
                                                        N_EDGES, IN_DIM);
    }
    {
        const int blocks = (N_NODES + 255) / 256;
        sage_invcnt_kernel<<<blocks, 256, 0, stream>>>(cnt, inv, N_NODES);
    }
    {
        const int mtiles = N_NODES / 16;                 // 6250 (exact)
        const int waves  = mtiles * (HID_DIM / 64);      // NT=4 -> 64 cols/wave
        const int blocks = (waves * 32 + 255) / 256;
        sage_gemm_kernel<IN_DIM, HID_DIM, 4, true>
            <<<blocks, 256, 0, stream>>>(agg1, inv, x, W1l, W1r, b1, h, N_NODES);
    }

    // Layer 2: scatter-mean of h, then out = mean@W2l^T + b2 + h@W2r^T
    {
        const long long total = (long long)N_EDGES * (HID_DIM / 4);
        const int blocks = (int)((total + 255) / 256);
        sage_scatter_kernel<<<blocks, 256, 0, stream>>>(h, src, dst, agg2, nullptr,
                                                        N_EDGES, HID_DIM);
    }
    {
        const int mtiles = N_NODES / 16;
        const int waves  = mtiles * (OUT_DIM / 64);
        const int blocks = (waves * 32 + 255) / 256;
        sage_gemm_kernel<HID_DIM, OUT_DIM, 4, false>
            <<<blocks, 256, 0, stream>>>(agg2, inv, h, W2l, W2r, b2, out, N_NODES);
    }
}